// PointNetSetAbstraction_85358180040863
// MI455X (gfx1250) — compile-verified
//
#include <hip/hip_runtime.h>
#include <hip/hip_bf16.h>

#define BB 16
#define NN 4096
#define SS 1024
#define KK 32
#define RAD2 0.04f
#define MM (BB * SS * KK)   // 524288 rows

typedef __attribute__((ext_vector_type(2))) float v2f;
typedef __attribute__((ext_vector_type(8))) float v8f;

typedef __attribute__((address_space(1))) int* gint_ptr;
typedef __attribute__((address_space(3))) int* lint_ptr;

#if __has_builtin(__builtin_amdgcn_global_load_async_to_lds_b32)
#define HAVE_ASYNC_LDS 1
#else
#define HAVE_ASYNC_LDS 0
#endif

__device__ __forceinline__ void async_fence_then_barrier() {
#if HAVE_ASYNC_LDS
#if __has_builtin(__builtin_amdgcn_s_wait_asynccnt)
    __builtin_amdgcn_s_wait_asynccnt(0);
#else
    asm volatile("s_wait_asynccnt 0x0" ::: "memory");
#endif
#endif
    __syncthreads();
}

// ---------------------------------------------------------------------------
// 1) Farthest point sampling: one block per batch, xyz cached in LDS.
//    Writes new_xyz_t [B,S,3] (scratch) and new_xyz output [B,3,S] (d_out).
// ---------------------------------------------------------------------------
__global__ __launch_bounds__(1024)
void fps_kernel(const float* __restrict__ xyz, float* __restrict__ new_xyz_t,
                float* __restrict__ out_xyz) {
    const int b    = blockIdx.x;
    const int tid  = threadIdx.x;
    const int lane = tid & 31;
    const int wid  = tid >> 5;

    __shared__ float sx[NN], sy[NN], sz[NN];
    __shared__ float rbest[32];
    __shared__ int   ridx[32];
    __shared__ float ccx, ccy, ccz;
    __shared__ int   cidx;

    const float* px = xyz + (size_t)b * 3 * NN;
    for (int n = tid; n < NN; n += 1024) {
        sx[n] = px[n];
        sy[n] = px[NN + n];
        sz[n] = px[2 * NN + n];
    }
    if (tid == 0) cidx = 0;

    float dist[NN / 1024];
#pragma unroll
    for (int i = 0; i < NN / 1024; ++i) dist[i] = 1e10f;
    __syncthreads();

    for (int it = 0; it < SS; ++it) {
        if (tid == 0) {
            const int c = cidx;
            const float cx = sx[c], cy = sy[c], cz = sz[c];
            ccx = cx; ccy = cy; ccz = cz;
            float* nxt = new_xyz_t + ((size_t)b * SS + it) * 3;
            nxt[0] = cx; nxt[1] = cy; nxt[2] = cz;
            float* ox = out_xyz + (size_t)b * 3 * SS;
            ox[it] = cx; ox[SS + it] = cy; ox[2 * SS + it] = cz;
        }
        __syncthreads();
        const float cx = ccx, cy = ccy, cz = ccz;

        float best = -1.0f;
        int   bi   = NN;
#pragma unroll
        for (int i = 0; i < NN / 1024; ++i) {
            const int n = tid + i * 1024;
            const float dx = sx[n] - cx, dy = sy[n] - cy, dz = sz[n] - cz;
            const float d  = dx * dx + dy * dy + dz * dz;
            const float dm = fminf(dist[i], d);
            dist[i] = dm;
            if (dm > best || (dm == best && n < bi)) { best = dm; bi = n; }
        }
        // wave32 argmax reduction (prefer lowest index on ties, like jnp.argmax)
#pragma unroll
        for (int off = 16; off; off >>= 1) {
            const float ob = __shfl_xor(best, off, 32);
            const int   oi = __shfl_xor(bi, off, 32);
            if (ob > best || (ob == best && oi < bi)) { best = ob; bi = oi; }
        }
        if (lane == 0) { rbest[wid] = best; ridx[wid] = bi; }
        __syncthreads();
        if (wid == 0) {
            best = rbest[lane]; bi = ridx[lane];
#pragma unroll
            for (int off = 16; off; off >>= 1) {
                const float ob = __shfl_xor(best, off, 32);
                const int   oi = __shfl_xor(bi, off, 32);
                if (ob > best || (ob == best && oi < bi)) { best = ob; bi = oi; }
            }
            if (lane == 0) cidx = bi;
        }
        __syncthreads();
    }
}

// ---------------------------------------------------------------------------
// 2) Ball query + gather: one wave32 per centroid. Ordered first-32 selection
//    via ballot/popcount prefix == reference's sort-by-index-take-32.
//    Builds X0 [M, 8] (6 channels + 2 zero pad) ready for WMMA.
// ---------------------------------------------------------------------------
__global__ __launch_bounds__(256)
void group_kernel(const float* __restrict__ xyz, const float* __restrict__ pts,
                  const float* __restrict__ new_xyz_t, float* __restrict__ X0) {
    const int lane = threadIdx.x & 31;
    const int wv   = threadIdx.x >> 5;
    const int g    = blockIdx.x * 8 + wv;     // centroid id in [0, B*S)
    const int b    = g >> 10;                 // g / SS

    __shared__ int grp[8][KK];
    grp[wv][lane] = 0;

    const float cx = new_xyz_t[(size_t)g * 3 + 0];
    const float cy = new_xyz_t[(size_t)g * 3 + 1];
    const float cz = new_xyz_t[(size_t)g * 3 + 2];
    const float* px = xyz + (size_t)b * 3 * NN;

    int count = 0;
    for (int base = 0; base < NN && count < KK; base += 32) {
        const int n = base + lane;
        const float dx = px[n] - cx, dy = px[NN + n] - cy, dz = px[2 * NN + n] - cz;
        const float d2 = dx * dx + dy * dy + dz * dz;
        const bool  in = (d2 <= RAD2);
        const unsigned mask = __builtin_amdgcn_ballot_w32(in);
        const int pos = count + __popc(mask & ((1u << lane) - 1u));
        if (in && pos < KK) grp[wv][pos] = n;
        count = min(count + __popc(mask), KK);
    }
    __syncthreads();
    if (lane >= count) grp[wv][lane] = grp[wv][0];   // pad with first neighbor
    __syncthreads();

    int idx = grp[wv][lane];
    idx = idx < 0 ? 0 : (idx >= NN ? NN - 1 : idx);
    const float* pp = pts + (size_t)b * 3 * NN;
    const float f0 = px[idx] - cx;
    const float f1 = px[NN + idx] - cy;
    const float f2 = px[2 * NN + idx] - cz;
    const float f3 = pp[idx];
    const float f4 = pp[NN + idx];
    const float f5 = pp[2 * NN + idx];

    float4* dst = (float4*)(X0 + ((size_t)g * KK + lane) * 8);
    dst[0] = make_float4(f0, f1, f2, f3);
    dst[1] = make_float4(f4, f5, 0.0f, 0.0f);
}

// ---------------------------------------------------------------------------
// 3) WMMA fp32 GEMM: Y[M,CO] = f(X)[M,CI] * W[CO,CIW]^T + bias, where
//    f = relu(x*scale[c]+shift[c]) of the PREVIOUS layer's BN when FUSE_BN.
//    One wave per 16-row strip; all CO tiles in accumulators; weights (and
//    BN scale/shift) staged in LDS, via async-to-LDS when available.
// ---------------------------------------------------------------------------
template <int CI, int CIW, int CO, bool FUSE_BN>
__global__ __launch_bounds__(256)
void gemm_kernel(const float* __restrict__ X, const float* __restrict__ W,
                 const float* __restrict__ bias, float* __restrict__ Y,
                 const float* __restrict__ bnst /* [scale[CI] | shift[CI]] */) {
    __shared__ float wlds[CO * CI + (FUSE_BN ? 2 * CI : 0)];
    for (int i = threadIdx.x; i < CO * CI; i += 256) {
        const int n = i / CI, k = i % CI;
#if HAVE_ASYNC_LDS
        if (k < CIW) {
            __builtin_amdgcn_global_load_async_to_lds_b32(
                (gint_ptr)(W + n * CIW + k), (lint_ptr)&wlds[i], 0, 0);
        } else {
            wlds[i] = 0.0f;
        }
#else
        wlds[i] = (k < CIW) ? W[n * CIW + k] : 0.0f;
#endif
    }
    if constexpr (FUSE_BN) {
        for (int i = threadIdx.x; i < 2 * CI; i += 256) {
#if HAVE_ASYNC_LDS
            __builtin_amdgcn_global_load_async_to_lds_b32(
                (gint_ptr)(bnst + i), (lint_ptr)&wlds[CO * CI + i], 0, 0);
#else
            wlds[CO * CI + i] = bnst[i];
#endif
        }
    }
    async_fence_then_barrier();

    const int lane = threadIdx.x & 31;
    const int wv   = threadIdx.x >> 5;
    const size_t rowbase = (size_t)blockIdx.x * 128 + (size_t)wv * 16;
    const int l15 = lane & 15;
    const int kh  = (lane >> 4) * 2;   // A/B fragment: lanes 16-31 hold K+2,K+3

    const float* xrow = X + (rowbase + l15) * CI + kh;
    __builtin_prefetch(X + (rowbase + 128) * CI, 0, 1);   // next block's rows

    v8f acc[CO / 16];
    const v8f zero8 = {0.f, 0.f, 0.f, 0.f, 0.f, 0.f, 0.f, 0.f};
#pragma unroll
    for (int t = 0; t < CO / 16; ++t) acc[t] = zero8;

    for (int k = 0; k < CI; k += 4) {
        v2f a = *(const v2f*)(xrow + k);
        if constexpr (FUSE_BN) {
            const v2f sc = *(const v2f*)&wlds[CO * CI + k + kh];
            const v2f sh = *(const v2f*)&wlds[CO * CI + CI + k + kh];
            a.x = fmaxf(fmaf(a.x, sc.x, sh.x), 0.0f);
            a.y = fmaxf(fmaf(a.y, sc.y, sh.y), 0.0f);
        }
#pragma unroll
        for (int t = 0; t < CO / 16; ++t) {
            const v2f bf = *(const v2f*)&wlds[(t * 16 + l15) * CI + k + kh];
            acc[t] = __builtin_amdgcn_wmma_f32_16x16x4_f32(
                false, a, false, bf, (short)0, acc[t], false, false);
        }
    }

    const size_t r0 = rowbase + (size_t)(lane >> 4) * 8;   // C/D: row = v + 8*(lane>=16)
#pragma unroll
    for (int t = 0; t < CO / 16; ++t) {
        const int n  = t * 16 + l15;
        const float bv = bias[n];
#pragma unroll
        for (int v = 0; v < 8; ++v) {
            Y[(r0 + v) * CO + n] = acc[t][v] + bv;
        }
    }
}

// ---------------------------------------------------------------------------
// 4) BN stats: zero, column sum/sumsq reduction (fp32 global atomics),
//    scale/shift prep. Stats layout: [sum CO | sumsq CO | scale CO | shift CO]
// ---------------------------------------------------------------------------
__global__ void zero_kernel(float* __restrict__ p, int n) {
    const int i = blockIdx.x * blockDim.x + threadIdx.x;
    if (i < n) p[i] = 0.0f;
}

template <int CO>
__global__ void colstats_kernel(const float* __restrict__ Y, float* __restrict__ stats) {
    const int c = threadIdx.x;                 // blockDim.x == CO
    const int rows_per_block = MM / gridDim.x;
    const size_t r0 = (size_t)blockIdx.x * rows_per_block;
    float s = 0.0f, sq = 0.0f;
    for (size_t r = r0; r < r0 + rows_per_block; ++r) {
        const float v = Y[r * CO + c];
        s += v;
        sq += v * v;
    }
    unsafeAtomicAdd(&stats[c], s);             // global_atomic_add_f32
    unsafeAtomicAdd(&stats[CO + c], sq);
}

template <int CO>
__global__ void bnprep_kernel(float* __restrict__ stats, const float* __restrict__ g,
                              const float* __restrict__ be) {
    const int c = threadIdx.x;
    const float invM = 1.0f / (float)MM;
    const float mean = stats[c] * invM;
    const float var  = stats[CO + c] * invM - mean * mean;   // biased var (train mode)
    const float sc   = g[c] * rsqrtf(var + 1e-5f);
    stats[2 * CO + c] = sc;
    stats[3 * CO + c] = be[c] - mean * sc;
}

// ---------------------------------------------------------------------------
// 5) Fused BN+ReLU+max over K -> new_points output [B,128,S] (after new_xyz).
// ---------------------------------------------------------------------------
__global__ __launch_bounds__(256)
void maxpool_kernel(const float* __restrict__ Y, const float* __restrict__ stats,
                    float* __restrict__ out) {
    const size_t i = (size_t)blockIdx.x * 256 + threadIdx.x;  // over B*S*128
    const int    c  = (int)(i & 127);
    const size_t bs = i >> 7;
    const float sc = stats[2 * 128 + c];
    const float sh = stats[3 * 128 + c];
    const float* p = Y + bs * KK * 128 + c;
    float m = 0.0f;   // relu output is >= 0, so 0 is a safe identity for max
#pragma unroll
    for (int k = 0; k < KK; ++k) {
        const float v = fmaxf(fmaf(p[(size_t)k * 128], sc, sh), 0.0f);
        m = fmaxf(m, v);
    }
    const int b = (int)(bs >> 10);
    const int s = (int)(bs & 1023);
    out[(size_t)BB * 3 * SS + ((size_t)b * 128 + c) * SS + s] = m;
}

// ---------------------------------------------------------------------------
extern "C" void kernel_launch(void* const* d_in, const int* in_sizes, int n_in,
                              void* d_out, int out_size, void* d_ws, size_t ws_size,
                              hipStream_t stream) {
    const float* xyz = (const float*)d_in[0];
    const float* pts = (const float*)d_in[1];
    const float* w0  = (const float*)d_in[2];
    const float* b0  = (const float*)d_in[3];
    const float* g0  = (const float*)d_in[4];
    const float* be0 = (const float*)d_in[5];
    const float* w1  = (const float*)d_in[6];
    const float* b1  = (const float*)d_in[7];
    const float* g1  = (const float*)d_in[8];
    const float* be1 = (const float*)d_in[9];
    const float* w2  = (const float*)d_in[10];
    const float* b2  = (const float*)d_in[11];
    const float* g2  = (const float*)d_in[12];
    const float* be2 = (const float*)d_in[13];
    float* out = (float*)d_out;

    char* ws = (char*)d_ws;
    float* new_xyz_t = (float*)(ws);                          // B*S*3   = 192 KB
    float* stats0    = (float*)(ws + ((size_t)1 << 18));      // 4*64  floats
    float* stats1    = (float*)(ws + ((size_t)1 << 18) + 2048);
    float* stats2    = (float*)(ws + ((size_t)1 << 18) + 4096);
    float* X0        = (float*)(ws + ((size_t)1 << 20));      // M*8   fp32 = 16 MB
    float* Y1        = (float*)(ws + ((size_t)32  << 20));    // M*64  fp32 = 128 MB
    float* Y2        = (float*)(ws + ((size_t)160 << 20));    // M*64  fp32 = 128 MB
    float* Y3        = (float*)(ws + ((size_t)288 << 20));    // M*128 fp32 = 256 MB

    fps_kernel<<<BB, 1024, 0, stream>>>(xyz, new_xyz_t, out);
    group_kernel<<<BB * SS / 8, 256, 0, stream>>>(xyz, pts, new_xyz_t, X0);

    // layer 0: 6 -> 64 (raw conv out in Y1; BN folded into layer-1 GEMM)
    gemm_kernel<8, 6, 64, false><<<MM / 128, 256, 0, stream>>>(X0, w0, b0, Y1, nullptr);
    zero_kernel<<<1, 256, 0, stream>>>(stats0, 4 * 64);
    colstats_kernel<64><<<1024, 64, 0, stream>>>(Y1, stats0);
    bnprep_kernel<64><<<1, 64, 0, stream>>>(stats0, g0, be0);

    // layer 1: 64 -> 64, consuming bn0(relu) on the fly
    gemm_kernel<64, 64, 64, true><<<MM / 128, 256, 0, stream>>>(Y1, w1, b1, Y2,
                                                                stats0 + 2 * 64);
    zero_kernel<<<1, 256, 0, stream>>>(stats1, 4 * 64);
    colstats_kernel<64><<<1024, 64, 0, stream>>>(Y2, stats1);
    bnprep_kernel<64><<<1, 64, 0, stream>>>(stats1, g1, be1);

    // layer 2: 64 -> 128, consuming bn1(relu) on the fly
    gemm_kernel<64, 64, 128, true><<<MM / 128, 256, 0, stream>>>(Y2, w2, b2, Y3,
                                                                 stats1 + 2 * 64);
    zero_kernel<<<2, 256, 0, stream>>>(stats2, 4 * 128);
    colstats_kernel<128><<<1024, 128, 0, stream>>>(Y3, stats2);
    bnprep_kernel<128><<<1, 128, 0, stream>>>(stats2, g2, be2);

    // fused bn2 + relu + max over K
    maxpool_kernel<<<BB * SS * 128 / 256, 256, 0, stream>>>(Y3, stats2, out);
}